// GraphAttentionNetwork_21010980012145
// MI455X (gfx1250) — compile-verified
//
#include <hip/hip_runtime.h>
#include <hip/hip_bf16.h>

typedef __attribute__((ext_vector_type(2))) float v2f;
typedef __attribute__((ext_vector_type(8))) float v8f;

#define LEAKY 0.2f
#define N_NODES 1024
#define ES 1032   // padded LDS row stride for score tile (breaks 256B bank period)

static __device__ __forceinline__ v8f wmma4(v2f a, v2f b, v8f c) {
  // V_WMMA_F32_16X16X4_F32 : D = A(16x4) * B(4x16) + C, fp32 exact
  return __builtin_amdgcn_wmma_f32_16x16x4_f32(false, a, false, b, (short)0, c,
                                               false, false);
}

static __device__ __forceinline__ float wave_max(float v) {
#pragma unroll
  for (int off = 16; off > 0; off >>= 1) v = fmaxf(v, __shfl_xor(v, off, 32));
  return v;
}
static __device__ __forceinline__ float wave_sum(float v) {
#pragma unroll
  for (int off = 16; off > 0; off >>= 1) v += __shfl_xor(v, off, 32);
  return v;
}

// ---------------------------------------------------------------------------
// C[M,N] = A[M,K] @ B[K,N], fp32 via V_WMMA_F32_16X16X4_F32.
// One wave per 16x16 C tile. M,K multiples of 4/16; N may be ragged (zero-pad).
// ---------------------------------------------------------------------------
__global__ __launch_bounds__(32) void gat_gemm_wmma_f32(
    const float* __restrict__ A, const float* __restrict__ B,
    float* __restrict__ C, int M, int N, int K) {
  const int tn = blockIdx.x * 16;
  const int tm = blockIdx.y * 16;
  const int lane = threadIdx.x & 31;
  const int half = lane >> 4;   // 0: K{0,1}, 1: K{2,3}
  const int ln   = lane & 15;   // A row / B col within tile

  const int arow = tm + ln;
  const int bcol = tn + ln;
  const int cok  = (bcol < N);

  v8f acc = {};
  for (int k0 = 0; k0 < K; k0 += 4) {
    const int ka = k0 + 2 * half;
    v2f af, bf;
    af.x = A[arow * K + ka];
    af.y = A[arow * K + ka + 1];
    const int bidx = cok ? (ka * N + bcol) : 0;     // clamped, stays in-bounds
    float b0 = B[bidx];
    float b1 = B[cok ? (bidx + N) : 0];
    bf.x = cok ? b0 : 0.0f;
    bf.y = cok ? b1 : 0.0f;
    acc = wmma4(af, bf, acc);
  }
#pragma unroll
  for (int r = 0; r < 8; ++r) {
    const int m = half * 8 + r;
    if (cok) C[(tm + m) * N + tn + ln] = acc[r];
  }
}

// ---------------------------------------------------------------------------
// Fused GATv2 attention, layer 1 (8 heads, f=32). One block = 16 target rows
// x one head. Scores -> masked softmax -> P@G (WMMA) -> ELU, all in LDS.
//   g : [1024, 8, 32]   adj : [1024,1024]   a : [32]   hout : [1024, 256]
// ---------------------------------------------------------------------------
__global__ __launch_bounds__(256) void gat_attn1_fused(
    const float* __restrict__ g, const int* __restrict__ adj,
    const float* __restrict__ a, float* __restrict__ hout) {
  extern __shared__ float smem[];
  float* e_s  = smem;                 // [16][ES]
  float* gi_s = e_s + 16 * ES;        // [16][32]
  float* sa   = gi_s + 16 * 32;       // [32]

  const int i0  = blockIdx.x * 16;
  const int h   = blockIdx.y;
  const int tid = threadIdx.x;
  const float NEG_INF = -__builtin_inff();

  // stage in target-row features + attention vector
  for (int idx = tid; idx < 16 * 32; idx += 256) {
    const int il = idx >> 5, f = idx & 31;
    gi_s[idx] = g[(i0 + il) * 256 + h * 32 + f];
  }
  if (tid < 32) sa[tid] = a[tid];
  __syncthreads();

  // --- scores: e[il][j] = sum_f a_f * lrelu(g_j + g_i), masked by adj -------
  for (int j = tid; j < N_NODES; j += 256) {
    float gj[32];
#pragma unroll
    for (int f = 0; f < 32; ++f) gj[f] = g[j * 256 + h * 32 + f];
#pragma unroll 4
    for (int il = 0; il < 16; ++il) {
      const int edge = adj[(i0 + il) * N_NODES + j];
      float acc = 0.0f;
#pragma unroll
      for (int f = 0; f < 32; ++f) {
        float v = gj[f] + gi_s[il * 32 + f];
        v = (v > 0.0f) ? v : LEAKY * v;
        acc = fmaf(sa[f], v, acc);
      }
      e_s[il * ES + j] = edge ? acc : NEG_INF;
    }
  }
  __syncthreads();

  // --- row softmax over j (wave w owns rows w and w+8) ----------------------
  {
    const int w = tid >> 5, lane = tid & 31;
#pragma unroll
    for (int rr = 0; rr < 2; ++rr) {
      float* er = e_s + (w + rr * 8) * ES;
      float m = NEG_INF;
      for (int j = lane; j < N_NODES; j += 32) m = fmaxf(m, er[j]);
      m = wave_max(m);
      float s = 0.0f;
      for (int j = lane; j < N_NODES; j += 32) {
        const float p = expf(er[j] - m);
        er[j] = p;
        s += p;
      }
      s = wave_sum(s);
      const float inv = 1.0f / s;
      for (int j = lane; j < N_NODES; j += 32) er[j] *= inv;
    }
  }
  __syncthreads();

  // --- out[16,32] = P[16,1024] @ G[1024,32] via WMMA; waves 0,1 -------------
  const int w = tid >> 5;
  if (w < 2) {                         // wave-uniform branch, EXEC stays full
    const int nt   = w;                // 16-col tile of f
    const int lane = tid & 31;
    const int half = lane >> 4;
    const int ln   = lane & 15;
    const int n    = nt * 16 + ln;
    v8f acc = {};
    for (int k0 = 0; k0 < N_NODES; k0 += 4) {
      const int ka = k0 + 2 * half;
      v2f af, bf;
      af.x = e_s[ln * ES + ka];
      af.y = e_s[ln * ES + ka + 1];
      bf.x = g[ka * 256 + h * 32 + n];
      bf.y = g[(ka + 1) * 256 + h * 32 + n];
      acc = wmma4(af, bf, acc);
    }
#pragma unroll
    for (int r = 0; r < 8; ++r) {
      const int m = half * 8 + r;
      const float v = acc[r];
      hout[(i0 + m) * 256 + h * 32 + n] = (v > 0.0f) ? v : expm1f(v);  // ELU
    }
  }
}

// ---------------------------------------------------------------------------
// Fused GATv2 attention, layer 2 (1 head, f=40, mean over 1 head == identity).
//   g : [1024, 40]   a : [40]   out : [1024, 40]
// ---------------------------------------------------------------------------
__global__ __launch_bounds__(256) void gat_attn2_fused(
    const float* __restrict__ g, const int* __restrict__ adj,
    const float* __restrict__ a, float* __restrict__ out) {
  extern __shared__ float smem[];
  float* e_s  = smem;                 // [16][ES]
  float* gi_s = e_s + 16 * ES;        // [16][40]
  float* sa   = gi_s + 16 * 40;       // [40]

  const int i0  = blockIdx.x * 16;
  const int tid = threadIdx.x;
  const float NEG_INF = -__builtin_inff();

  for (int idx = tid; idx < 16 * 40; idx += 256) {
    const int il = idx / 40, f = idx % 40;
    gi_s[idx] = g[(i0 + il) * 40 + f];
  }
  if (tid < 40) sa[tid] = a[tid];
  __syncthreads();

  for (int j = tid; j < N_NODES; j += 256) {
    float gj[40];
#pragma unroll
    for (int f = 0; f < 40; ++f) gj[f] = g[j * 40 + f];
#pragma unroll 4
    for (int il = 0; il < 16; ++il) {
      const int edge = adj[(i0 + il) * N_NODES + j];
      float acc = 0.0f;
#pragma unroll
      for (int f = 0; f < 40; ++f) {
        float v = gj[f] + gi_s[il * 40 + f];
        v = (v > 0.0f) ? v : LEAKY * v;
        acc = fmaf(sa[f], v, acc);
      }
      e_s[il * ES + j] = edge ? acc : NEG_INF;
    }
  }
  __syncthreads();

  {
    const int w = tid >> 5, lane = tid & 31;
#pragma unroll
    for (int rr = 0; rr < 2; ++rr) {
      float* er = e_s + (w + rr * 8) * ES;
      float m = NEG_INF;
      for (int j = lane; j < N_NODES; j += 32) m = fmaxf(m, er[j]);
      m = wave_max(m);
      float s = 0.0f;
      for (int j = lane; j < N_NODES; j += 32) {
        const float p = expf(er[j] - m);
        er[j] = p;
        s += p;
      }
      s = wave_sum(s);
      const float inv = 1.0f / s;
      for (int j = lane; j < N_NODES; j += 32) er[j] *= inv;
    }
  }
  __syncthreads();

  const int w = tid >> 5;
  if (w < 3) {                         // 3 n-tiles cover f=40 (last ragged)
    const int nt   = w;
    const int lane = tid & 31;
    const int half = lane >> 4;
    const int ln   = lane & 15;
    const int n    = nt * 16 + ln;
    const int nok  = (n < 40);
    v8f acc = {};
    for (int k0 = 0; k0 < N_NODES; k0 += 4) {
      const int ka = k0 + 2 * half;
      v2f af, bf;
      af.x = e_s[ln * ES + ka];
      af.y = e_s[ln * ES + ka + 1];
      const int bidx = nok ? (ka * 40 + n) : 0;
      float b0 = g[bidx];
      float b1 = g[nok ? (bidx + 40) : 0];
      bf.x = nok ? b0 : 0.0f;
      bf.y = nok ? b1 : 0.0f;
      acc = wmma4(af, bf, acc);
    }
#pragma unroll
    for (int r = 0; r < 8; ++r) {
      const int m = half * 8 + r;
      if (nok) out[(i0 + m) * 40 + n] = acc[r];
    }
  }
}

extern "C" void kernel_launch(void* const* d_in, const int* in_sizes, int n_in,
                              void* d_out, int out_size, void* d_ws, size_t ws_size,
                              hipStream_t stream) {
  const float* x   = (const float*)d_in[0];   // [1024, 256]
  const int*   adj = (const int*)  d_in[1];   // [1024, 1024]
  const float* W1  = (const float*)d_in[2];   // [256, 256]
  const float* a1  = (const float*)d_in[3];   // [32]
  const float* W2  = (const float*)d_in[4];   // [256, 40]
  const float* a2  = (const float*)d_in[5];   // [40]
  float* outp = (float*)d_out;                // [1024, 40]

  float* g1 = (float*)d_ws;                   // [1024, 256]
  float* hb = g1 + 1024 * 256;                // [1024, 256]
  float* g2 = hb + 1024 * 256;                // [1024, 40]

  // g1 = x @ W1
  {
    dim3 grid(256 / 16, 1024 / 16);
    gat_gemm_wmma_f32<<<grid, 32, 0, stream>>>(x, W1, g1, 1024, 256, 256);
  }
  // fused attention layer 1 (+ELU) -> hb
  {
    const size_t smem = (16 * ES + 16 * 32 + 32) * sizeof(float);
    dim3 grid(1024 / 16, 8);
    gat_attn1_fused<<<grid, 256, smem, stream>>>(g1, adj, a1, hb);
  }
  // g2 = hb @ W2
  {
    dim3 grid((40 + 15) / 16, 1024 / 16);
    gat_gemm_wmma_f32<<<grid, 32, 0, stream>>>(hb, W2, g2, 1024, 40, 256);
  }
  // fused attention layer 2 -> out
  {
    const size_t smem = (16 * ES + 16 * 40 + 40) * sizeof(float);
    dim3 grid(1024 / 16);
    gat_attn2_fused<<<grid, 256, smem, stream>>>(g2, adj, a2, outp);
  }
}